// GATLayer_30416958390449
// MI455X (gfx1250) — compile-verified
//
#include <hip/hip_runtime.h>
#include <hip/hip_bf16.h>

#define NN    4096
#define EE    131072
#define INF   512
#define OUTF  64
#define HH    8
#define SLOPE 0.2f
#define BITW  (NN / 32)   // 128 words per row

typedef __attribute__((ext_vector_type(2))) float v2f;
typedef __attribute__((ext_vector_type(8))) float v8f;

// ---------------- zero helper ----------------
__global__ void zero_u32(unsigned int* p, int n) {
    int i = blockIdx.x * blockDim.x + threadIdx.x;
    if (i < n) p[i] = 0u;
}

// ---------------- edge dedup: bitmask ----------------
__global__ void mark_edges(const long long* __restrict__ ei, unsigned int* __restrict__ bits) {
    int e = blockIdx.x * blockDim.x + threadIdx.x;
    if (e >= EE) return;
    int i = (int)ei[e];        // src
    int j = (int)ei[EE + e];   // dst
    atomicOr(&bits[i * BITW + (j >> 5)], 1u << (j & 31));
}

// ---------------- degree = popcount of bitmask row ----------------
__global__ void node_deg(const unsigned int* __restrict__ bits, unsigned int* __restrict__ deg) {
    int i = blockIdx.x * blockDim.x + threadIdx.x;
    if (i >= NN) return;
    const unsigned int* row = bits + i * BITW;
    unsigned int d = 0;
    for (int w = 0; w < BITW; ++w) d += __popc(row[w]);
    deg[i] = d;
}

// ---------------- exclusive scan over 4096 degrees (1 block, 1024 thr, 4/thr) ----------------
__global__ void scan_deg(const unsigned int* __restrict__ deg, unsigned int* __restrict__ rowstart) {
    __shared__ unsigned int sums[1024];
    int t = threadIdx.x;
    unsigned int d0 = deg[4 * t], d1 = deg[4 * t + 1], d2 = deg[4 * t + 2], d3 = deg[4 * t + 3];
    unsigned int s = d0 + d1 + d2 + d3;
    sums[t] = s;
    __syncthreads();
    for (int off = 1; off < 1024; off <<= 1) {
        unsigned int v = (t >= off) ? sums[t - off] : 0u;
        __syncthreads();
        sums[t] += v;
        __syncthreads();
    }
    unsigned int excl = sums[t] - s;   // exclusive prefix for this thread's group
    rowstart[4 * t]     = excl;
    rowstart[4 * t + 1] = excl + d0;
    rowstart[4 * t + 2] = excl + d0 + d1;
    rowstart[4 * t + 3] = excl + d0 + d1 + d2;
    if (t == 1023) rowstart[NN] = sums[1023];
}

// ---------------- CSR fill from bitmask (sorted, deduped) ----------------
__global__ void fill_csr(const unsigned int* __restrict__ bits,
                         const unsigned int* __restrict__ rowstart,
                         unsigned int* __restrict__ csr) {
    int i = blockIdx.x * blockDim.x + threadIdx.x;
    if (i >= NN) return;
    unsigned int base = rowstart[i], cnt = 0;
    const unsigned int* row = bits + i * BITW;
    for (int w = 0; w < BITW; ++w) {
        unsigned int b = row[w];
        while (b) {
            int j = __ffs(b) - 1;
            csr[base + cnt++] = (unsigned int)(w * 32 + j);
            b &= b - 1;
        }
    }
}

// ---------------- xh[h,n,o] = sum_k W[h,o,k] * x[n,k] via V_WMMA_F32_16X16X4_F32 ----------------
// One wave per (head, 16-node tile): 4 accumulators cover OUTF=64, 128 K-steps of 4.
__global__ void xh_wmma(const float* __restrict__ x, const float* __restrict__ W,
                        float* __restrict__ xh) {
    int wid  = (blockIdx.x * blockDim.x + threadIdx.x) >> 5;  // 0..2047 (exact)
    int lane = threadIdx.x & 31;
    int h  = wid & 7;
    int nt = wid >> 3;            // 0..255 node tiles
    int row = lane & 15;          // A: M index / B: N (out-feature) index
    int kh  = lane >> 4;          // K half select (K = 2*kh + vgpr)
    const float* xrow = x + (size_t)(nt * 16 + row) * INF;
    const float* Wh   = W + (size_t)h * OUTF * INF;
    v8f acc[4] = {};
    for (int k0 = 0; k0 < INF; k0 += 4) {
        v2f a;
        a.x = xrow[k0 + 2 * kh];
        a.y = xrow[k0 + 2 * kh + 1];
#pragma unroll
        for (int t = 0; t < 4; ++t) {
            const float* wrow = Wh + (size_t)(t * 16 + row) * INF;  // B[k][o] = W[h][o][k]
            v2f b;
            b.x = wrow[k0 + 2 * kh];
            b.y = wrow[k0 + 2 * kh + 1];
            acc[t] = __builtin_amdgcn_wmma_f32_16x16x4_f32(
                false, a, false, b, (short)0, acc[t], false, false);
        }
    }
    // C/D layout: VGPR r -> M = r + 8*(lane>=16); N = lane&15
    float* ob = xh + ((size_t)h * NN + (size_t)nt * 16) * OUTF;
#pragma unroll
    for (int r = 0; r < 8; ++r) {
        int M = r + 8 * kh;
#pragma unroll
        for (int t = 0; t < 4; ++t)
            ob[(size_t)M * OUTF + t * 16 + row] = acc[t][r];
    }
}

// ---------------- attention logits per node: ssrc (a2), sdst (a1) ----------------
__global__ void s_kernel(const float* __restrict__ xh, const float* __restrict__ a,
                         float* __restrict__ ssrc, float* __restrict__ sdst) {
    int idx = blockIdx.x * blockDim.x + threadIdx.x;   // over H*N
    if (idx >= HH * NN) return;
    int h = idx >> 12, n = idx & (NN - 1);
    const float* xr = xh + ((size_t)h * NN + n) * OUTF;
    const float* a1 = a + h * 2 * OUTF;        // column (dst) term
    const float* a2 = a1 + OUTF;               // row (src) term
    float sj = 0.f, si = 0.f;
    for (int o = 0; o < OUTF; ++o) {
        float v = xr[o];
        sj += v * a1[o];
        si += v * a2[o];
    }
    sdst[h * NN + n] = sj;
    ssrc[h * NN + n] = si;
}

// ---------------- S[h,o] = sum_n xh[h,n,o] (grid HHx16, atomic partials) ----------------
__global__ void colsum(const float* __restrict__ xh, float* __restrict__ S) {
    int h = blockIdx.x, seg = blockIdx.y;
    int o = threadIdx.x & 63, sub = threadIdx.x >> 6;
    const float* p = xh + ((size_t)h * NN + seg * 256 + sub * 64) * OUTF + o;
    float s = 0.f;
    for (int k = 0; k < 64; ++k) s += p[k * OUTF];
    __shared__ float red[256];
    red[threadIdx.x] = s;
    __syncthreads();
    if (sub == 0) {
        s += red[threadIdx.x + 64] + red[threadIdx.x + 128] + red[threadIdx.x + 192];
        atomicAdd(&S[h * OUTF + o], s);
    }
}

// ---------------- one wave per (h, i): softmax stats + gather + rank-1 background ----------------
__global__ void row_attn(const unsigned int* __restrict__ rowstart,
                         const unsigned int* __restrict__ csr,
                         const float* __restrict__ xh,
                         const float* __restrict__ ssrc, const float* __restrict__ sdst,
                         const float* __restrict__ S, float* __restrict__ out) {
    int wid  = (blockIdx.x * blockDim.x + threadIdx.x) >> 5;
    int lane = threadIdx.x & 31;
    if (wid >= HH * NN) return;
    int h = wid >> 12, i = wid & (NN - 1);
    unsigned int rs = rowstart[i], re = rowstart[i + 1];
    float si = ssrc[h * NN + i];
    // pass 1: row max (background logit 0 always present since deg < N)
    float m = 0.f;
    for (unsigned int p = rs + lane; p < re; p += 32) {
        float e = si + sdst[h * NN + (int)csr[p]];
        float l = (e >= 0.f) ? e : SLOPE * e;
        m = fmaxf(m, l);
    }
    for (int o = 16; o > 0; o >>= 1) m = fmaxf(m, __shfl_xor(m, o, 32));
    // pass 2: Z
    float zs = 0.f;
    for (unsigned int p = rs + lane; p < re; p += 32) {
        float e = si + sdst[h * NN + (int)csr[p]];
        float l = (e >= 0.f) ? e : SLOPE * e;
        zs += __expf(l - m);
    }
    for (int o = 16; o > 0; o >>= 1) zs += __shfl_xor(zs, o, 32);
    float bg  = __expf(-m);
    unsigned int deg = re - rs;
    float inv = 1.f / (zs + (float)(NN - deg) * bg);
    // pass 3: out = sum_nbrs (w - bg)/Z * xh[j]  +  bg/Z * S
    float acc0 = 0.f, acc1 = 0.f;
    const float* xhh = xh + (size_t)h * NN * OUTF;
    for (unsigned int p0 = rs; p0 < re; p0 += 32) {
        unsigned int p = p0 + lane;
        int j = 0; float w = 0.f;
        if (p < re) {
            j = (int)csr[p];
            float e = si + sdst[h * NN + j];
            float l = (e >= 0.f) ? e : SLOPE * e;
            w = __expf(l - m) - bg;
        }
        int cnt = (int)min(32u, re - p0);
        for (int b = 0; b < cnt; ++b) {
            float coeff = __shfl(w, b, 32) * inv;
            int   jj    = __shfl(j, b, 32);
            const float* xr = xhh + (size_t)jj * OUTF;
            acc0 += coeff * xr[lane];
            acc1 += coeff * xr[lane + 32];
        }
    }
    float c0 = bg * inv;
    acc0 += c0 * S[h * OUTF + lane];
    acc1 += c0 * S[h * OUTF + lane + 32];
    out[(size_t)i * (HH * OUTF) + h * OUTF + lane]      = acc0;
    out[(size_t)i * (HH * OUTF) + h * OUTF + lane + 32] = acc1;
}

extern "C" void kernel_launch(void* const* d_in, const int* in_sizes, int n_in,
                              void* d_out, int out_size, void* d_ws, size_t ws_size,
                              hipStream_t stream) {
    const float*     x  = (const float*)d_in[0];
    const long long* ei = (const long long*)d_in[1];
    const float*     W  = (const float*)d_in[2];
    const float*     a  = (const float*)d_in[3];
    float* out = (float*)d_out;

    char* ws = (char*)d_ws;
    size_t off = 0;
    auto take = [&](size_t bytes) { size_t o = off; off += (bytes + 255) & ~(size_t)255; return o; };
    unsigned int* bits     = (unsigned int*)(ws + take((size_t)NN * BITW * 4));
    unsigned int* deg      = (unsigned int*)(ws + take((size_t)NN * 4));
    unsigned int* rowstart = (unsigned int*)(ws + take((size_t)(NN + 1) * 4));
    unsigned int* csr      = (unsigned int*)(ws + take((size_t)EE * 4));
    float*        xh       = (float*)(ws + take((size_t)HH * NN * OUTF * 4));
    float*        sdst     = (float*)(ws + take((size_t)HH * NN * 4));
    float*        ssrc     = (float*)(ws + take((size_t)HH * NN * 4));
    float*        Ssum     = (float*)(ws + take((size_t)HH * OUTF * 4));

    // 1. zero bitmask + S
    zero_u32<<<(NN * BITW + 255) / 256, 256, 0, stream>>>(bits, NN * BITW);
    zero_u32<<<(HH * OUTF + 255) / 256, 256, 0, stream>>>((unsigned int*)Ssum, HH * OUTF);
    // 2. dedup edges into bitmask
    mark_edges<<<(EE + 255) / 256, 256, 0, stream>>>(ei, bits);
    // 3. degrees, scan, CSR
    node_deg<<<(NN + 255) / 256, 256, 0, stream>>>(bits, deg);
    scan_deg<<<1, 1024, 0, stream>>>(deg, rowstart);
    fill_csr<<<(NN + 255) / 256, 256, 0, stream>>>(bits, rowstart, csr);
    // 4. fp32 WMMA feature transform: 2048 waves = 256 blocks x 256 threads
    xh_wmma<<<256, 256, 0, stream>>>(x, W, xh);
    // 5. per-node attention logits, per-head column sums
    s_kernel<<<(HH * NN + 255) / 256, 256, 0, stream>>>(xh, a, ssrc, sdst);
    colsum<<<dim3(HH, 16), 256, 0, stream>>>(xh, Ssum);
    // 6. softmax + sparse gather + rank-1 background: one wave per (h,i)
    row_attn<<<(HH * NN * 32) / 256, 256, 0, stream>>>(rowstart, csr, xh, ssrc, sdst, Ssum, out);
}